// SoftHeatMap_31808527794314
// MI455X (gfx1250) — compile-verified
//
#include <hip/hip_runtime.h>
#include <cstdint>

#define MOUNT  63
#define TILE_X 16
#define TILE_Y 64
#define YPT    4      // y-pixels per thread; 16 x * 16 y-groups = 256 threads

__global__ __launch_bounds__(256)
void softheatmap_kernel(const float* __restrict__ boxes,
                        const float* __restrict__ mount,
                        float* __restrict__ out) {
    __shared__ float  s_lut[64 * 64];   // 16 KB gaussian LUT
    __shared__ int4   s_box[128];       // compacted: x0,y0,w,h
    __shared__ float2 s_inv[128];       // 1/w, 1/h
    __shared__ int    s_cnt;

    const int t = threadIdx.x;
    const int tile_x0 = blockIdx.x * TILE_X;
    const int tile_y0 = blockIdx.y * TILE_Y;

    if (t == 0) s_cnt = 0;

    // ---- CDNA5 async data mover: stage 16KB LUT global -> LDS ----
    // GVS form: global addr = SADDR(64b) + VADDR(32b offset); VDST holds LDS byte addr.
    {
        unsigned lds_base = (unsigned)(uintptr_t)(&s_lut[0]);   // low 32 bits of generic LDS ptr == DS offset
        unsigned long long gbase = (unsigned long long)(uintptr_t)mount;
#pragma unroll
        for (int k = 0; k < 4; ++k) {
            unsigned byte_off = (unsigned)((t + k * 256) * 16);
            unsigned lds_addr = lds_base + byte_off;
            asm volatile("global_load_async_to_lds_b128 %0, %1, %2"
                         :: "v"(lds_addr), "v"(byte_off), "s"(gbase)
                         : "memory");
        }
    }

    __syncthreads();   // s_cnt init visible to all waves

    // ---- per-tile box culling + param precompute (exact fp32 match of reference) ----
    if (t < 128) {
        float4 b = reinterpret_cast<const float4*>(boxes)[t];   // cx, cy, bw, bh
        int x0 = (int)((b.x - 0.5f * b.z) * 1024.0f);
        int x1 = (int)((b.x + 0.5f * b.z) * 1024.0f);
        int y0 = (int)((b.y - 0.5f * b.w) * 1024.0f);
        int y1 = (int)((b.y + 0.5f * b.w) * 1024.0f);
        int w = x1 - x0, h = y1 - y0;
        bool hit = (w > 0) && (h > 0) &&
                   (x0 < tile_x0 + TILE_X) && (x0 + w > tile_x0) &&
                   (y0 < tile_y0 + TILE_Y) && (y0 + h > tile_y0);
        if (hit) {
            int idx = atomicAdd(&s_cnt, 1);                      // ds_add_rtn_u32
            s_box[idx] = make_int4(x0, y0, w, h);
            s_inv[idx] = make_float2(1.0f / (float)w, 1.0f / (float)h);
        }
    }

    asm volatile("s_wait_asynccnt 0" ::: "memory");   // async LDS writes complete (this wave)
    __syncthreads();                                   // ...and all waves' writes visible

    const int n = s_cnt;
    const int x = tile_x0 + (t >> 4);
    const int y = tile_y0 + (t & 15) * YPT;

    float m[YPT] = {0.f, 0.f, 0.f, 0.f};

    for (int i = 0; i < n; ++i) {
        int4   bb = s_box[i];
        float2 iv = s_inv[i];

        int  rx = x - bb.x;
        bool vx = (rx >= 0) & (rx < bb.z);
        int  tr = rx * MOUNT;
        int  row = (int)((float)tr * iv.x);       // ~floor(tr/w), error < 1
        row += ((row + 1) * bb.z <= tr);          // exact floor fixup
        row -= (row * bb.z > tr);
        row = min(max(row, 0), MOUNT - 1);        // clip like reference (v_med3)
        const float* lr = &s_lut[row * 64];

#pragma unroll
        for (int j = 0; j < YPT; ++j) {
            int  ry = (y + j) - bb.y;
            bool v  = vx & (ry >= 0) & (ry < bb.w);
            int  tc = ry * MOUNT;
            int  col = (int)((float)tc * iv.y);
            col += ((col + 1) * bb.w <= tc);
            col -= (col * bb.w > tc);
            col = min(max(col, 0), MOUNT - 1);
            float g = lr[col];                     // ds_load_b32 gather
            m[j] = v ? fmaxf(m[j], g) : m[j];
        }
    }

    // coalesced float4 store: img[x, y..y+3] -> out[x*1024 + y]
    *reinterpret_cast<float4*>(&out[x * 1024 + y]) = make_float4(m[0], m[1], m[2], m[3]);
}

extern "C" void kernel_launch(void* const* d_in, const int* in_sizes, int n_in,
                              void* d_out, int out_size, void* d_ws, size_t ws_size,
                              hipStream_t stream) {
    const float* boxes = (const float*)d_in[0];   // 128 x 4 fp32
    const float* mount = (const float*)d_in[1];   // 64 x 64 fp32
    float* out = (float*)d_out;                   // 1024 x 1024 fp32

    dim3 grid(1024 / TILE_X, 1024 / TILE_Y);      // 64 x 16 = 1024 blocks
    softheatmap_kernel<<<grid, 256, 0, stream>>>(boxes, mount, out);
}